// UniformBottomUpHTMM_3444563771783
// MI455X (gfx1250) — compile-verified
//
#include <hip/hip_runtime.h>
#include <hip/hip_bf16.h>

// ---------------------------------------------------------------------------
// UniformBottomUpHTMM on MI455X (gfx1250).
//   - one wave32 workgroup per (tree, ensemble-member g): 512*8 = 4096 blocks
//   - tree recursion fully LDS-resident (ping-pong level buffers, ~29 KB/WG;
//     WGP has 320 KB)
//   - einsum  t[e,i] = sum_j smA[i,j] * beta[e,j]  via V_WMMA_F32_16X16X4_F32
//     (full fp32), 2 WMMAs per 16-edge batch, A-operand software-pipelined
//   - softmax reparameterization hoisted to a tiny prelude kernel -> d_ws
// ---------------------------------------------------------------------------

#define HT_NTREES 512
#define HT_DEPTH  10
#define HT_C      8
#define HT_M      128
#define HT_G      8
#define HT_NPT    1023   // 2^10 - 1 nodes per tree

typedef __attribute__((ext_vector_type(2))) float v2f;
typedef __attribute__((ext_vector_type(8))) float v8f;

// ---------------------------------------------------------------------------
// Prelude: softmax(A, axis=0), softmax(B, axis=1), softmax(Pi, axis=0).
// A:(8,8,8) idx i*64+j*8+g ; B:(8,128,8) idx c*1024+m*8+g ; Pi:(8,8) c*8+g
// ---------------------------------------------------------------------------
__global__ void htmm_softmax_kernel(const float* __restrict__ A,
                                    const float* __restrict__ B,
                                    const float* __restrict__ Pi,
                                    float* __restrict__ smA,
                                    float* __restrict__ smB,
                                    float* __restrict__ smPi) {
  const int t = threadIdx.x;  // one block of 256 threads
  if (t < 64) {               // smA: softmax over i for each (j,g)
    const int j = t >> 3, g = t & 7;
    float v[8], mx = -1e30f, s = 0.f;
    for (int i = 0; i < 8; ++i) { v[i] = A[i * 64 + j * 8 + g]; mx = fmaxf(mx, v[i]); }
    for (int i = 0; i < 8; ++i) { v[i] = __expf(v[i] - mx); s += v[i]; }
    const float inv = 1.f / s;
    for (int i = 0; i < 8; ++i) smA[i * 64 + j * 8 + g] = v[i] * inv;
  } else if (t < 128) {       // smB: softmax over m for each (c,g)
    const int u = t - 64, c = u >> 3, g = u & 7;
    float mx = -1e30f, s = 0.f;
    for (int m = 0; m < HT_M; ++m) mx = fmaxf(mx, B[c * 1024 + m * 8 + g]);
    for (int m = 0; m < HT_M; ++m) s += __expf(B[c * 1024 + m * 8 + g] - mx);
    const float inv = 1.f / s;
    for (int m = 0; m < HT_M; ++m)
      smB[c * 1024 + m * 8 + g] = __expf(B[c * 1024 + m * 8 + g] - mx) * inv;
  } else if (t < 136) {       // smPi: softmax over c for each g
    const int g = t - 128;
    float v[8], mx = -1e30f, s = 0.f;
    for (int c = 0; c < 8; ++c) { v[c] = Pi[c * 8 + g]; mx = fmaxf(mx, v[c]); }
    for (int c = 0; c < 8; ++c) { v[c] = __expf(v[c] - mx); s += v[c]; }
    const float inv = 1.f / s;
    for (int c = 0; c < 8; ++c) smPi[c * 8 + g] = v[c] * inv;
  }
}

// ---------------------------------------------------------------------------
// Main: one wave per (tree, g). Upward recursion with WMMA.
// ---------------------------------------------------------------------------
__global__ __launch_bounds__(32)
void htmm_upward_kernel(const float* __restrict__ smA,
                        const float* __restrict__ smB,
                        const float* __restrict__ smPi,
                        const int*   __restrict__ x,
                        float*       __restrict__ out) {
  __shared__ float sB[HT_C * HT_M];   // emission slice for this g : sB[c*128+m]
  __shared__ float sA[HT_C * HT_C];   // transition slice         : sA[i*8+j]
  __shared__ float sPi[HT_C];
  __shared__ float bufA[512 * 8];     // level ping buffer (beta rows, 32B each)
  __shared__ float bufB[256 * 8];     // level pong buffer

  const int tree  = (int)blockIdx.x >> 3;
  const int g     = (int)blockIdx.x & 7;
  const int lane  = (int)threadIdx.x;  // 0..31, single wave32
  const int xbase = tree * HT_NPT;

  // prefetch this tree's symbol slice (~4 KB): one cacheline per lane
  __builtin_prefetch(&x[xbase + lane * 32], 0, 1);

  // --- load per-g parameters into LDS ---
  for (int t = lane; t < HT_C * HT_M; t += 32) {
    const int c = t >> 7, m = t & 127;
    sB[t] = smB[c * 1024 + m * 8 + g];
  }
  {
    int t = lane;            // sA[i*8+j] = smA[i,j,g]
    sA[t] = smA[(t >> 3) * 64 + (t & 7) * 8 + g];
    t = lane + 32;
    sA[t] = smA[(t >> 3) * 64 + (t & 7) * 8 + g];
  }
  if (lane < HT_C) sPi[lane] = smPi[lane * 8 + g];
  __syncthreads();

  const int n  = lane & 15;   // WMMA column index (parent state i when n<8)
  const int hi = lane >> 4;   // lane half

  // B-operand (smA_g^T) held in registers for the whole tree.
  // f32 4x16 B layout: VGPR0 = {K0 | K2}, VGPR1 = {K1 | K3}, N = lane%16.
  v2f bop0 = {0.f, 0.f}, bop1 = {0.f, 0.f};
  if (n < 8) {
    bop0.x = sA[n * 8 + (hi ? 2 : 0)];        // B[K=j, N=i] = smA[i,j]
    bop0.y = sA[n * 8 + (hi ? 3 : 1)];
    bop1.x = sA[n * 8 + 4 + (hi ? 2 : 0)];
    bop1.y = sA[n * 8 + 4 + (hi ? 3 : 1)];
  }

  // --- leaves (depth 9, 512 nodes -> bufA) ---
  float ll = 0.f;
  for (int k = lane; k < 512; k += 32) {
    const int m = x[xbase + 511 + k];
    float v[8], nu = 0.f;
#pragma unroll
    for (int c = 0; c < 8; ++c) { v[c] = sPi[c] * sB[c * 128 + m]; nu += v[c]; }
    const float inv = 1.f / nu;
    float4* wp = (float4*)&bufA[k * 8];          // 32B-aligned row
    wp[0] = make_float4(v[0] * inv, v[1] * inv, v[2] * inv, v[3] * inv);
    wp[1] = make_float4(v[4] * inv, v[5] * inv, v[6] * inv, v[7] * inv);
    ll += __logf(nu);
  }
  __syncthreads();

  // --- upward recursion: child level S -> parent level S/2 ---
  float* child  = bufA;
  float* parent = bufB;
  int S = 512, pdepth = 8;
  while (S >= 2) {
    const int P = S >> 1;

    // ---- phase 1: t_beta via WMMA (16 child edges / batch) ----
    if (S >= 16) {
      // software-pipelined A-operand: two b64 LDS loads per lane per batch
      v2f a0, a1;
      {
        const float* rowp = &child[n * 8];
        a0.x = rowp[2 * hi];     a0.y = rowp[2 * hi + 1];
        a1.x = rowp[4 + 2 * hi]; a1.y = rowp[4 + 2 * hi + 1];
      }
      for (int m0 = 0; m0 < S; m0 += 16) {
        const v2f c0 = a0, c1 = a1;
        if (m0 + 16 < S) {  // prefetch next batch while this one computes
          const float* rowp = &child[(m0 + 16 + n) * 8];
          a0.x = rowp[2 * hi];     a0.y = rowp[2 * hi + 1];
          a1.x = rowp[4 + 2 * hi]; a1.y = rowp[4 + 2 * hi + 1];
        }
        v8f acc = {};
        acc = __builtin_amdgcn_wmma_f32_16x16x4_f32(false, c0, false, bop0,
                                                    (short)0, acc, false, false);
        acc = __builtin_amdgcn_wmma_f32_16x16x4_f32(false, c1, false, bop1,
                                                    (short)0, acc, false, false);
        // D[M=edge, N=state]; children 2q,2q+1 -> parent q (scatter-mean, cnt=2)
        if (n < 8) {
          float* pp = &parent[((m0 >> 1) + 4 * hi) * 8 + n];
#pragma unroll
          for (int q = 0; q < 4; ++q)
            pp[q * 8] = 0.5f * (acc[2 * q] + acc[2 * q + 1]);
        }
      }
    } else {
      // tail levels S in {8,4,2}: one partial batch, guarded stores
      const float* rowp = &child[n * 8];
      v2f c0, c1;
      c0.x = rowp[2 * hi];     c0.y = rowp[2 * hi + 1];
      c1.x = rowp[4 + 2 * hi]; c1.y = rowp[4 + 2 * hi + 1];
      v8f acc = {};
      acc = __builtin_amdgcn_wmma_f32_16x16x4_f32(false, c0, false, bop0,
                                                  (short)0, acc, false, false);
      acc = __builtin_amdgcn_wmma_f32_16x16x4_f32(false, c1, false, bop1,
                                                  (short)0, acc, false, false);
      if (n < 8) {
#pragma unroll
        for (int q = 0; q < 4; ++q) {
          const int p = 4 * hi + q;
          if (p < P) parent[p * 8 + n] = 0.5f * (acc[2 * q] + acc[2 * q + 1]);
        }
      }
    }
    __syncthreads();

    // ---- phase 2: beta = normalize(t_beta * B[:, x[parent]]) ; ll += log(nu)
    const int nodebase = xbase + (1 << pdepth) - 1;
    for (int p = lane; p < P; p += 32) {
      const int m = x[nodebase + p];
      float4* rp = (float4*)&parent[p * 8];
      float4 t0 = rp[0], t1 = rp[1];
      float v[8];
      v[0] = t0.x * sB[0 * 128 + m]; v[1] = t0.y * sB[1 * 128 + m];
      v[2] = t0.z * sB[2 * 128 + m]; v[3] = t0.w * sB[3 * 128 + m];
      v[4] = t1.x * sB[4 * 128 + m]; v[5] = t1.y * sB[5 * 128 + m];
      v[6] = t1.z * sB[6 * 128 + m]; v[7] = t1.w * sB[7 * 128 + m];
      float nu = 0.f;
#pragma unroll
      for (int c = 0; c < 8; ++c) nu += v[c];
      const float inv = 1.f / nu;
      rp[0] = make_float4(v[0] * inv, v[1] * inv, v[2] * inv, v[3] * inv);
      rp[1] = make_float4(v[4] * inv, v[5] * inv, v[6] * inv, v[7] * inv);
      ll += __logf(nu);
    }
    __syncthreads();

    float* tmp = child; child = parent; parent = tmp;
    S = P; --pdepth;
  }

  // --- reduce ll across the wave, write out[tree, g] ---
#pragma unroll
  for (int off = 16; off > 0; off >>= 1) ll += __shfl_xor(ll, off, 32);
  if (lane == 0) out[tree * HT_G + g] = ll;
}

// ---------------------------------------------------------------------------
extern "C" void kernel_launch(void* const* d_in, const int* in_sizes, int n_in,
                              void* d_out, int out_size, void* d_ws, size_t ws_size,
                              hipStream_t stream) {
  (void)in_sizes; (void)n_in; (void)out_size; (void)ws_size;
  const float* A  = (const float*)d_in[0];   // (8,8,8)
  const float* B  = (const float*)d_in[1];   // (8,128,8)
  const float* Pi = (const float*)d_in[2];   // (8,8)
  const int*   x  = (const int*)d_in[3];     // (512*1023,)

  float* smA  = (float*)d_ws;        // 512 floats
  float* smB  = smA + 512;           // 8192 floats
  float* smPi = smB + 8192;          // 64 floats

  htmm_softmax_kernel<<<1, 256, 0, stream>>>(A, B, Pi, smA, smB, smPi);
  htmm_upward_kernel<<<HT_NTREES * HT_G, 32, 0, stream>>>(smA, smB, smPi, x,
                                                          (float*)d_out);
}